// GraphGATConvBn_10866267259206
// MI455X (gfx1250) — compile-verified
//
#include <hip/hip_runtime.h>
#include <hip/hip_bf16.h>

#ifndef __has_builtin
#define __has_builtin(x) 0
#endif

#define N_NODES   50000
#define N_EDGES   800000
#define N_TOTAL   (N_EDGES + N_NODES)   // edges + self loops
#define CH        128                    // IN_C == OUT_C
#define MTILES    (N_NODES / 16)         // 3125 exactly
#define NEG_SLOPE 0.2f
#define BN_EPS    1e-5f
#define MI        2                      // M-tiles per wave (amortize B-fragment ds loads)

typedef __attribute__((ext_vector_type(2)))  float  v2f;
typedef __attribute__((ext_vector_type(8)))  float  v8f;
typedef __attribute__((ext_vector_type(4)))  int    v4i;

// CDNA5 async global->LDS path (ASYNCcnt), with safe fallback
#if defined(__gfx1250__) && \
    __has_builtin(__builtin_amdgcn_global_load_async_to_lds_b128) && \
    __has_builtin(__builtin_amdgcn_s_wait_asynccnt)
#define USE_ASYNC_LDS 1
typedef __attribute__((address_space(1))) v4i* gp_v4i;   // global v4i*
typedef __attribute__((address_space(3))) v4i* lp_v4i;   // LDS v4i*
#else
#define USE_ASYNC_LDS 0
#endif

// ---------------------------------------------------------------- init
__global__ __launch_bounds__(256)
void k_init(float* __restrict__ outacc, unsigned* __restrict__ umax,
            float* __restrict__ denom, float* __restrict__ gsum,
            float* __restrict__ gsumsq) {
  int i = blockIdx.x * blockDim.x + threadIdx.x;
  if (i < N_NODES * CH) outacc[i] = 0.0f;
  if (i < N_NODES) { umax[i] = 0u; denom[i] = 0.0f; }
  if (i < CH)     { gsum[i] = 0.0f; gsumsq[i] = 0.0f; }
}

// ---------------------------------------------------------------- GEMM h = x @ W  (f32 WMMA)
__global__ __launch_bounds__(128)
void k_gemm(const float* __restrict__ x, const float* __restrict__ W,
            float* __restrict__ h) {
  __shared__ float Wl[CH * CH];               // 64 KB of the 320 KB WGP LDS
  const int tid = threadIdx.x;

  // Stage W once per block.
#if USE_ASYNC_LDS
  {
    const float4* gw = (const float4*)W;
    float4* lw = (float4*)Wl;
    for (int i = tid; i < CH * CH / 4; i += 128) {
      __builtin_amdgcn_global_load_async_to_lds_b128(
          (gp_v4i)(gw + i), (lp_v4i)(lw + i),
          0, 0);                               // emits global_load_async_to_lds_b128
    }
    __builtin_amdgcn_s_wait_asynccnt(0);       // ASYNCcnt == 0 -> LDS writes landed
  }
#else
  {
    const float4* gw = (const float4*)W;
    float4* lw = (float4*)Wl;
    for (int i = tid; i < CH * CH / 4; i += 128) lw[i] = gw[i];
  }
#endif
  __syncthreads();

  const int wave = tid >> 5;
  const int lane = tid & 31;
  const int hi   = lane >> 4;                  // lane half (0/1)
  const int ln   = lane & 15;
  const int mt0  = blockIdx.x * (4 * MI) + wave * MI;   // first 16-row tile of this wave

  bool valid[MI];
  int  srow[MI];
  #pragma unroll
  for (int mi = 0; mi < MI; ++mi) {
    valid[mi] = (mt0 + mi) < MTILES;
    srow[mi]  = valid[mi] ? (mt0 + mi) * 16 + ln : ln;  // clamp to a safe row
  }
  if (!valid[0]) return;                       // whole-wave uniform

  v8f acc[MI][8];
  #pragma unroll
  for (int mi = 0; mi < MI; ++mi)
    #pragma unroll
    for (int i = 0; i < 8; ++i) acc[mi][i] = (v8f)0.0f;

  for (int kb = 0; kb < CH; kb += 4) {
    // A 16x4 f32 frag: elem v -> k = kb + v + 2*hi (contiguous pair per lane)
    v2f a[MI];
    #pragma unroll
    for (int mi = 0; mi < MI; ++mi)
      a[mi] = *(const v2f*)(x + srow[mi] * CH + kb + 2 * hi);
    #pragma unroll
    for (int nt = 0; nt < 8; ++nt) {
      const int col = nt * 16 + ln;
      v2f b;                                   // B 4x16 frag: elem v -> k = kb + v + 2*hi
      b.x = Wl[(kb + 2 * hi    ) * CH + col];
      b.y = Wl[(kb + 2 * hi + 1) * CH + col];
      #pragma unroll
      for (int mi = 0; mi < MI; ++mi)
        acc[mi][nt] = __builtin_amdgcn_wmma_f32_16x16x4_f32(
            false, a[mi], false, b, (short)0, acc[mi][nt], false, false);
    }
  }

  // C/D layout: VGPR r -> m = r + 8*hi, n = lane%16
  #pragma unroll
  for (int mi = 0; mi < MI; ++mi) {
    if (!valid[mi]) continue;
    #pragma unroll
    for (int nt = 0; nt < 8; ++nt)
      #pragma unroll
      for (int r = 0; r < 8; ++r)
        h[((mt0 + mi) * 16 + r + 8 * hi) * CH + nt * 16 + ln] = acc[mi][nt][r];
  }
}

// ---------------------------------------------------------------- per-node attention logits
__global__ __launch_bounds__(256)
void k_att(const float* __restrict__ h, const float* __restrict__ att_src,
           const float* __restrict__ att_dst, float* __restrict__ a_s,
           float* __restrict__ a_d) {
  const int lane = threadIdx.x & 31;
  const int node = blockIdx.x * 8 + (threadIdx.x >> 5);  // one wave32 per node
  if (node >= N_NODES) return;
  const float4 hv = ((const float4*)(h + (size_t)node * CH))[lane];
  const float4 as = ((const float4*)att_src)[lane];
  const float4 ad = ((const float4*)att_dst)[lane];
  float s = hv.x * as.x + hv.y * as.y + hv.z * as.z + hv.w * as.w;
  float d = hv.x * ad.x + hv.y * ad.y + hv.z * ad.z + hv.w * ad.w;
  #pragma unroll
  for (int off = 16; off > 0; off >>= 1) {
    s += __shfl_xor(s, off);
    d += __shfl_xor(d, off);
  }
  if (lane == 0) { a_s[node] = s; a_d[node] = d; }
}

__device__ __forceinline__ void edge_sd(const int* ei, int i, int& s, int& d) {
  if (i < N_EDGES) { s = ei[i]; d = ei[N_EDGES + i]; }
  else             { s = d = i - N_EDGES; }            // self loop
}
__device__ __forceinline__ float lrelu(float v) { return fmaxf(v, NEG_SLOPE * v); }

// ---------------------------------------------------------------- pass A: segment max
__global__ __launch_bounds__(256)
void k_edge_max(const int* __restrict__ ei, const float* __restrict__ a_s,
                const float* __restrict__ a_d, unsigned* __restrict__ umax) {
  const int i = blockIdx.x * blockDim.x + threadIdx.x;
  if (i >= N_TOTAL) return;
  int s, d; edge_sd(ei, i, s, d);
  const float e = lrelu(a_s[s] + a_d[d]);
  unsigned u = __float_as_uint(e);               // order-preserving f32 -> u32
  u = (u & 0x80000000u) ? ~u : (u | 0x80000000u);
  atomicMax(umax + d, u);
}

// ---------------------------------------------------------------- pass B: exp + segment sum
__global__ __launch_bounds__(256)
void k_edge_expsum(const int* __restrict__ ei, const float* __restrict__ a_s,
                   const float* __restrict__ a_d, const unsigned* __restrict__ umax,
                   float* __restrict__ denom, float* __restrict__ exbuf) {
  const int i = blockIdx.x * blockDim.x + threadIdx.x;
  if (i >= N_TOTAL) return;
  int s, d; edge_sd(ei, i, s, d);
  const float e = lrelu(a_s[s] + a_d[d]);
  const unsigned u = umax[d];
  const float m = (u & 0x80000000u) ? __uint_as_float(u ^ 0x80000000u)
                                    : __uint_as_float(~u);
  const float ex = __expf(e - m);                // e - m <= 0
  exbuf[i] = ex;
  atomicAdd(denom + d, ex);
}

// ---------------------------------------------------------------- pass C: weighted scatter-add (the hot loop)
__global__ __launch_bounds__(256)
void k_edge_agg(const int* __restrict__ ei, const float* __restrict__ exbuf,
                const float* __restrict__ denom, const float* __restrict__ h,
                float* __restrict__ outacc) {
  const int gid  = blockIdx.x * blockDim.x + threadIdx.x;
  const int item = gid >> 5;                     // one wave32 per edge
  const int lane = gid & 31;
  if (item >= N_TOTAL) return;
  int s, d; edge_sd(ei, item, s, d);
  const float alpha = exbuf[item] / denom[d];
  // lane owns 4 contiguous channels: one global_load_b128 + 4 f32 atomics
  const float4 hv = ((const float4*)(h + (size_t)s * CH))[lane];
  float* od = outacc + (size_t)d * CH + lane * 4;
  atomicAdd(od + 0, alpha * hv.x);
  atomicAdd(od + 1, alpha * hv.y);
  atomicAdd(od + 2, alpha * hv.z);
  atomicAdd(od + 3, alpha * hv.w);
}

// ---------------------------------------------------------------- BN stats (per-channel sum / sumsq)
__global__ __launch_bounds__(256)
void k_bn_stats(const float* __restrict__ outacc, float* __restrict__ gsum,
                float* __restrict__ gsumsq) {
  __shared__ float ssum[256], ssq[256];
  const int c    = threadIdx.x & (CH - 1);
  const int half = threadIdx.x >> 7;             // 0/1
  const int base = blockIdx.x * 128;
  float s = 0.0f, q = 0.0f;
  for (int r = base + half; r < base + 128 && r < N_NODES; r += 2) {
    const float v = outacc[(size_t)r * CH + c];
    s += v; q += v * v;
  }
  ssum[threadIdx.x] = s; ssq[threadIdx.x] = q;
  __syncthreads();
  if (half == 0) {
    s += ssum[threadIdx.x + 128];
    q += ssq[threadIdx.x + 128];
    atomicAdd(gsum + c, s);
    atomicAdd(gsumsq + c, q);
  }
}

// ---------------------------------------------------------------- BN finalize (scale/shift per channel)
__global__ void k_bn_final(const float* __restrict__ gsum, const float* __restrict__ gsumsq,
                           const float* __restrict__ gamma, const float* __restrict__ beta,
                           float* __restrict__ scale, float* __restrict__ shift) {
  const int c = threadIdx.x;
  if (c >= CH) return;
  const float inv  = 1.0f / (float)N_NODES;
  const float mean = gsum[c] * inv;
  const float var  = gsumsq[c] * inv - mean * mean;  // biased variance
  const float rs   = rsqrtf(var + BN_EPS);
  const float sc   = rs * gamma[c];
  scale[c] = sc;
  shift[c] = beta[c] - mean * sc;
  // NB: the GAT bias adds a per-channel constant to every node, which shifts
  // mean identically and therefore cancels exactly in (out - mean): skipped.
}

// ---------------------------------------------------------------- BN apply + ReLU (in place in d_out)
__global__ __launch_bounds__(256)
void k_bn_apply(float* __restrict__ outacc, const float* __restrict__ scale,
                const float* __restrict__ shift) {
  const int i = blockIdx.x * blockDim.x + threadIdx.x;
  if (i >= N_NODES * CH) return;
  const int c = i & (CH - 1);
  outacc[i] = fmaxf(outacc[i] * scale[c] + shift[c], 0.0f);
}

// ---------------------------------------------------------------- launch
extern "C" void kernel_launch(void* const* d_in, const int* in_sizes, int n_in,
                              void* d_out, int out_size, void* d_ws, size_t ws_size,
                              hipStream_t stream) {
  const float* x       = (const float*)d_in[0];
  const int*   ei      = (const int*)  d_in[1];   // edge_index [2, E] (row0=src, row1=dst)
  // d_in[2] = edge_attr (ignored by reference: edge_dim=None)
  const float* W       = (const float*)d_in[3];
  const float* att_src = (const float*)d_in[4];
  const float* att_dst = (const float*)d_in[5];
  // d_in[6] = bias: cancels through batch-norm (see k_bn_final)
  const float* gamma   = (const float*)d_in[7];
  const float* beta    = (const float*)d_in[8];

  float* outacc = (float*)d_out;                  // f32 accumulator, normalized in place

  // workspace carve-out (~29.8 MB)
  char* p = (char*)d_ws;
  float*    h      = (float*)p;    p += (size_t)N_NODES * CH * sizeof(float);
  float*    a_s    = (float*)p;    p += (size_t)N_NODES * sizeof(float);
  float*    a_d    = (float*)p;    p += (size_t)N_NODES * sizeof(float);
  unsigned* umax   = (unsigned*)p; p += (size_t)N_NODES * sizeof(unsigned);
  float*    denom  = (float*)p;    p += (size_t)N_NODES * sizeof(float);
  float*    exbuf  = (float*)p;    p += (size_t)N_TOTAL * sizeof(float);
  float*    gsum   = (float*)p;    p += CH * sizeof(float);
  float*    gsumsq = (float*)p;    p += CH * sizeof(float);
  float*    scale  = (float*)p;    p += CH * sizeof(float);
  float*    shift  = (float*)p;    p += CH * sizeof(float);

  const int elemBlocks = (N_NODES * CH + 255) / 256;          // 25000
  const int edgeBlocks = (N_TOTAL + 255) / 256;               // 3321
  const int aggBlocks  = (N_TOTAL * 32 + 255) / 256;          // 106250
  const int gemmBlocks = (MTILES + 4 * MI - 1) / (4 * MI);    // 391 (8 M-tiles/block)

  k_init<<<elemBlocks, 256, 0, stream>>>(outacc, umax, denom, gsum, gsumsq);
  k_gemm<<<gemmBlocks, 128, 0, stream>>>(x, W, h);
  k_att<<<(N_NODES + 7) / 8, 256, 0, stream>>>(h, att_src, att_dst, a_s, a_d);
  k_edge_max<<<edgeBlocks, 256, 0, stream>>>(ei, a_s, a_d, umax);
  k_edge_expsum<<<edgeBlocks, 256, 0, stream>>>(ei, a_s, a_d, umax, denom, exbuf);
  k_edge_agg<<<aggBlocks, 256, 0, stream>>>(ei, exbuf, denom, h, outacc);
  k_bn_stats<<<(N_NODES + 127) / 128, 256, 0, stream>>>(outacc, gsum, gsumsq);
  k_bn_final<<<1, 128, 0, stream>>>(gsum, gsumsq, gamma, beta, scale, shift);
  k_bn_apply<<<elemBlocks, 256, 0, stream>>>(outacc, scale, shift);
}